// NodeNet_11123965297189
// MI455X (gfx1250) — compile-verified
//
#include <hip/hip_runtime.h>
#include <math.h>

typedef float v2f __attribute__((ext_vector_type(2)));
typedef float v8f __attribute__((ext_vector_type(8)));

#define NQ     15
#define DIM    32768      // 2^15
#define NNODE  128
#define NEDGE  1024
#define NGATES 50

// ---------------------------------------------------------------------------
// Kernel A: bo[j,k] = sum_m Ro[m,j]*X[m,k];  bi[j,k] = sum_m Ri[m,j]*X[m,k]
// 1024 threads total (one per edge j). Loads of Ro/Ri are coalesced across j.
// ---------------------------------------------------------------------------
__global__ __launch_bounds__(256) void prep_kernel(
    const float* __restrict__ X,   // [128,5]
    const float* __restrict__ Ri,  // [128,1024]
    const float* __restrict__ Ro,  // [128,1024]
    float* __restrict__ bo,        // [1024,5]
    float* __restrict__ bi)        // [1024,5]
{
    const int j = blockIdx.x * blockDim.x + threadIdx.x;
    if (j >= NEDGE) return;
    float ao[5] = {0.f, 0.f, 0.f, 0.f, 0.f};
    float ai[5] = {0.f, 0.f, 0.f, 0.f, 0.f};
    for (int m = 0; m < NNODE; ++m) {
        const float ro = Ro[m * NEDGE + j];
        const float ri = Ri[m * NEDGE + j];
        #pragma unroll
        for (int k = 0; k < 5; ++k) {
            const float x = X[m * 5 + k];
            ao[k] += ro * x;
            ai[k] += ri * x;
        }
    }
    #pragma unroll
    for (int k = 0; k < 5; ++k) {
        bo[j * 5 + k] = ao[k];
        bi[j * 5 + k] = ai[k];
    }
}

// ---------------------------------------------------------------------------
// Kernel B: WMMA f32 16x16x4.
//   wave 0: mi = (Ri * e) @ bo  -> M[:, 0:5]
//   wave 1: mo = (Ro * e) @ bi  -> M[:, 5:10]
// grid.x = 8 row tiles of 16; block = 64 threads (2 wave32s).
// A frag: lane l holds A[M = l%16][K = 2*(l/16)+r], r=0,1  (ISA 16x4 f32 layout)
// B frag: lane l holds B[K = 2*(l/16)+r][N = l%16]
// D:      VGPR r -> row (r + 8*(l/16)), col l%16
// B columns >= 5 are zero-padded via clamped-address load + mask multiply so
// the inner loop stays branch-free (EXEC all-ones throughout, as WMMA needs).
// Also copies X into M[:, 10:15].
// ---------------------------------------------------------------------------
__global__ __launch_bounds__(64) void graph_wmma_kernel(
    const float* __restrict__ X,   // [128,5]
    const float* __restrict__ e,   // [1024]
    const float* __restrict__ Ri,  // [128,1024]
    const float* __restrict__ Ro,  // [128,1024]
    const float* __restrict__ bo,  // [1024,5]
    const float* __restrict__ bi,  // [1024,5]
    float* __restrict__ M)         // [128,15]
{
    const int tile = blockIdx.x;            // rows tile*16 .. tile*16+15
    const int wave = threadIdx.x >> 5;      // 0 -> mi, 1 -> mo
    const int lane = threadIdx.x & 31;
    const int col  = lane & 15;
    const int half = lane >> 4;

    const float* __restrict__ R  = wave ? Ro : Ri;
    const float* __restrict__ Bm = wave ? bi : bo;

    const int   rowA  = tile * 16 + col;    // A-matrix row for this lane
    const int   colc  = (col < 5) ? col : 0;   // clamped (always in-bounds)
    const float bmask = (col < 5) ? 1.0f : 0.0f;
    v8f acc = {};

    for (int k0 = 0; k0 < NEDGE; k0 += 4) {
        const int ka = k0 + 2 * half;
        v2f a, b;
        a.x = R[rowA * NEDGE + ka]     * e[ka];
        a.y = R[rowA * NEDGE + ka + 1] * e[ka + 1];
        b.x = Bm[ka * 5 + colc]       * bmask;   // unconditional load + mask
        b.y = Bm[(ka + 1) * 5 + colc] * bmask;
        acc = __builtin_amdgcn_wmma_f32_16x16x4_f32(
            /*neg_a=*/false, a, /*neg_b=*/false, b,
            /*c_mod=*/(short)0, acc, /*reuse_a=*/false, /*reuse_b=*/false);
    }

    if (col < 5) {
        #pragma unroll
        for (int r = 0; r < 8; ++r) {
            const int row = tile * 16 + r + 8 * half;
            M[row * 15 + wave * 5 + col] = acc[r];
        }
    }
    // M[:,10:15] = X
    if (wave == 0 && lane < 16) {
        const int row = tile * 16 + lane;
        #pragma unroll
        for (int k = 0; k < 5; ++k) M[row * 15 + 10 + k] = X[row * 5 + k];
    }
}

// ---------------------------------------------------------------------------
// Kernel C: statevector simulation, one node per workgroup.
// State (128 KB) lives entirely in LDS (320 KB/WGP on CDNA5).
// qubit w <-> bit (14 - w) of the flat amplitude index.
// ---------------------------------------------------------------------------
__global__ __launch_bounds__(1024) void circuit_kernel(
    const float* __restrict__ Mang,   // [128,15]
    const float* __restrict__ theta,  // [31]
    float* __restrict__ out)          // [128,2]
{
    __shared__ float st[DIM];
    __shared__ float red[64];

    const int n = blockIdx.x;
    const int t = threadIdx.x;

    // --- initial product state: RY(M[n,w]) on |0..0> for every qubit -------
    float cw[NQ], sw[NQ];
    #pragma unroll
    for (int w = 0; w < NQ; ++w) {
        const float h = 0.5f * Mang[n * NQ + w];
        cw[w] = cosf(h);
        sw[w] = sinf(h);
    }
    #pragma unroll 4
    for (int j = 0; j < 32; ++j) {
        const int idx = t + 1024 * j;
        float amp = 1.0f;
        #pragma unroll
        for (int pos = 0; pos < NQ; ++pos) {          // bit pos <-> qubit 14-pos
            amp *= ((idx >> pos) & 1) ? sw[14 - pos] : cw[14 - pos];
        }
        st[idx] = amp;
    }
    __syncthreads();

    // --- gate list (compile-time constant, fully unrolled) ------------------
    constexpr int g_op[NGATES] = {0,0,1,0,0,1,0,0,1,0,0,1,0,0,1,0,0,1,0,0,1,0,0,0,1,
                                  0,0,1,0,0,1,0,0,1,0,0,1,0,0,1,0,0,0,0,1,0,0,1,0,0};
    constexpr int g_a[NGATES]  = {0,1,0,2,3,3,4,5,4,6,7,7,8,9,8,10,11,11,12,13,8,14,15,16,1,
                                  14,15,6,16,17,9,18,19,9,19,20,2,21,22,13,23,24,25,26,0,27,28,14,29,30};
    constexpr int g_b[NGATES]  = {0,1,1,2,3,2,4,5,5,6,7,6,8,9,9,10,11,10,12,13,9,14,1,2,2,
                                  5,6,5,9,10,10,13,14,10,2,5,5,10,13,10,5,10,0,5,5,10,14,10,5,10};

    #pragma unroll
    for (int g = 0; g < NGATES; ++g) {
        if (g_op[g] == 0) {
            // RY(theta[a]) on qubit b  ->  bit position bp = 14 - b
            const int bp = 14 - g_b[g];
            const float h = 0.5f * theta[g_a[g]];
            const float c = cosf(h);
            const float s = sinf(h);
            const int lowmask = (1 << bp) - 1;
            #pragma unroll 4
            for (int j = 0; j < 16; ++j) {            // 16384 pairs / 1024 thr
                const int p  = t + 1024 * j;
                const int i0 = ((p >> bp) << (bp + 1)) | (p & lowmask);
                const int i1 = i0 | (1 << bp);
                const float a0 = st[i0];
                const float a1 = st[i1];
                st[i0] = c * a0 - s * a1;
                st[i1] = s * a0 + c * a1;
            }
        } else {
            // CNOT control a, target b -> swap amps with ctrl=1 over target bit
            const int bc = 14 - g_a[g];
            const int bt = 14 - g_b[g];
            const int lo = (bc < bt) ? bc : bt;
            const int hi = (bc < bt) ? bt : bc;
            #pragma unroll 4
            for (int j = 0; j < 8; ++j) {             // 8192 swaps / 1024 thr
                const int p    = t + 1024 * j;        // 13 free bits
                const int low  = p & ((1 << lo) - 1);
                const int mid  = (p >> lo) & ((1 << (hi - lo - 1)) - 1);
                const int high = p >> (hi - 1);
                int i0 = (high << (hi + 1)) | (mid << (lo + 1)) | low;
                i0 |= (1 << bc);                      // control = 1, target = 0
                const int i1 = i0 | (1 << bt);
                const float a0 = st[i0];
                const float a1 = st[i1];
                st[i0] = a1;
                st[i1] = a0;
            }
        }
        __syncthreads();
    }

    // --- <Z_5> (bit 9) and <Z_10> (bit 4) -----------------------------------
    float z5 = 0.f, z10 = 0.f;
    #pragma unroll 4
    for (int j = 0; j < 32; ++j) {
        const int idx = t + 1024 * j;
        const float a  = st[idx];
        const float a2 = a * a;
        z5  += ((idx >> 9) & 1) ? -a2 : a2;
        z10 += ((idx >> 4) & 1) ? -a2 : a2;
    }
    #pragma unroll
    for (int off = 16; off > 0; off >>= 1) {
        z5  += __shfl_xor(z5,  off, 32);
        z10 += __shfl_xor(z10, off, 32);
    }
    const int wid = t >> 5;
    if ((t & 31) == 0) { red[wid] = z5; red[32 + wid] = z10; }
    __syncthreads();
    if (t == 0) {
        float s5 = 0.f, s10 = 0.f;
        for (int i = 0; i < 32; ++i) { s5 += red[i]; s10 += red[32 + i]; }
        const float pi = 3.14159265358979323846f;
        out[n * 2 + 0] = pi * (1.0f - s5);
        out[n * 2 + 1] = pi * (1.0f - s10);
    }
}

// ---------------------------------------------------------------------------
extern "C" void kernel_launch(void* const* d_in, const int* in_sizes, int n_in,
                              void* d_out, int out_size, void* d_ws, size_t ws_size,
                              hipStream_t stream) {
    const float* X     = (const float*)d_in[0];   // [128,5]
    const float* e     = (const float*)d_in[1];   // [1024]
    const float* Ri    = (const float*)d_in[2];   // [128,1024]
    const float* Ro    = (const float*)d_in[3];   // [128,1024]
    const float* theta = (const float*)d_in[4];   // [31]
    float* out = (float*)d_out;                   // [128,2]

    float* ws = (float*)d_ws;
    float* bo = ws;                               // 1024*5
    float* bi = ws + NEDGE * 5;                   // 1024*5
    float* M  = ws + 2 * NEDGE * 5;               // 128*15

    prep_kernel<<<4, 256, 0, stream>>>(X, Ri, Ro, bo, bi);
    graph_wmma_kernel<<<8, 64, 0, stream>>>(X, e, Ri, Ro, bo, bi, M);
    circuit_kernel<<<NNODE, 1024, 0, stream>>>(M, theta, out);
}